// LiftedEmbeddingLoss_82927228551478
// MI455X (gfx1250) — compile-verified
//
#include <hip/hip_runtime.h>
#include <math.h>

typedef __attribute__((ext_vector_type(2))) float v2f;
typedef __attribute__((ext_vector_type(8))) float v8f;

#define KPAIRS 64
#define DIM    128
#define MARGIN 1.0f

// One wave (32 lanes) per triplet. For each 16-row tile of the K=64 pairs we
// compute the Gram row-sums Σ_d (a-p)^2 and Σ_d (a-n)^2 with fp32 WMMA
// (A = squared-diff tile, B = ones), which broadcasts the 16 distances across
// all output columns: lane half h holds distances for tile rows 8h..8h+7 in
// c[0..7]. These stream into an online logsumexp.
__global__ __launch_bounds__(256)
void lifted_loss_kernel(const float* __restrict__ emb,
                        const long long* __restrict__ trip,
                        float* __restrict__ losses, int T) {
  const int lane = threadIdx.x & 31;
  const int wave = threadIdx.x >> 5;
  const int t = blockIdx.x * 8 + wave;
  if (t >= T) return;                 // wave-uniform: EXEC stays all-ones

  const int r = lane & 15;            // row within 16-row tile (A-matrix M)
  const int h = lane >> 4;            // half selects dim pair within K-chunk

  v2f ones; ones.x = 1.0f; ones.y = 1.0f;   // B = 4x16 all-ones (layout-free)

  float m_ap = -__builtin_inff(), s_ap = 0.0f;
  float m_an = -__builtin_inff(), s_an = 0.0f;

  const long long* tb = trip + (long long)t * 3 * KPAIRS;

  for (int tau = 0; tau < 4; ++tau) {
    const int k = tau * 16 + r;
    const long long ia  = tb[k];
    const long long ip  = tb[KPAIRS + k];
    const long long inx = tb[2 * KPAIRS + k];
    const float* Ar = emb + ia  * DIM;
    const float* Pr = emb + ip  * DIM;
    const float* Nr = emb + inx * DIM;

    v8f c_ap = {};
    v8f c_an = {};
#pragma unroll
    for (int kc = 0; kc < 32; ++kc) {
      const int off = kc * 4 + h * 2;             // A-layout: half h owns dims 4kc+2h,+1
      v2f av = *(const v2f*)(Ar + off);
      v2f pv = *(const v2f*)(Pr + off);
      v2f nv = *(const v2f*)(Nr + off);
      v2f dap = av - pv;
      v2f dan = av - nv;
      v2f eap = dap * dap;
      v2f ean = dan * dan;
      // (neg_a, A, neg_b, B, c_mod, C, reuse_a, reuse_b)
      c_ap = __builtin_amdgcn_wmma_f32_16x16x4_f32(false, eap, false, ones,
                                                   (short)0, c_ap, false, false);
      c_an = __builtin_amdgcn_wmma_f32_16x16x4_f32(false, ean, false, ones,
                                                   (short)0, c_an, false, false);
    }

    // Online logsumexp over this tile's 8 distances per lane.
    float mx = c_ap[0];
    float my = MARGIN - c_an[0];
#pragma unroll
    for (int j = 1; j < 8; ++j) {
      mx = fmaxf(mx, c_ap[j]);
      my = fmaxf(my, MARGIN - c_an[j]);
    }
    float nm = fmaxf(m_ap, mx);
    float ns = s_ap * __expf(m_ap - nm);
#pragma unroll
    for (int j = 0; j < 8; ++j) ns += __expf(c_ap[j] - nm);
    m_ap = nm; s_ap = ns;

    nm = fmaxf(m_an, my);
    ns = s_an * __expf(m_an - nm);
#pragma unroll
    for (int j = 0; j < 8; ++j) ns += __expf((MARGIN - c_an[j]) - nm);
    m_an = nm; s_an = ns;
  }

  // Merge the two halves of the wave (rows 0-7 vs 8-15 of each tile).
  {
    float om = __shfl_xor(m_ap, 16, 32);
    float os = __shfl_xor(s_ap, 16, 32);
    float nm = fmaxf(m_ap, om);
    s_ap = s_ap * __expf(m_ap - nm) + os * __expf(om - nm);
    m_ap = nm;
  }
  {
    float om = __shfl_xor(m_an, 16, 32);
    float os = __shfl_xor(s_an, 16, 32);
    float nm = fmaxf(m_an, om);
    s_an = s_an * __expf(m_an - nm) + os * __expf(om - nm);
    m_an = nm;
  }

  if (lane == 0) {
    const float ap = m_ap + __logf(s_ap);
    const float an = m_an + __logf(s_an);
    const float l = ap + an;
    losses[t] = l > 0.0f ? l : 0.0f;
  }
}

// Deterministic fixed-order tree reduction of the per-triplet losses.
__global__ __launch_bounds__(256)
void lifted_loss_reduce(const float* __restrict__ losses,
                        float* __restrict__ out, int T) {
  __shared__ float sm[256];
  float acc = 0.0f;
  for (int i = threadIdx.x; i < T; i += 256) acc += losses[i];
  sm[threadIdx.x] = acc;
  __syncthreads();
  for (int s = 128; s > 0; s >>= 1) {
    if ((int)threadIdx.x < s) sm[threadIdx.x] += sm[threadIdx.x + s];
    __syncthreads();
  }
  if (threadIdx.x == 0) {
    out[0] = sm[0];
    out[1] = (float)T;   // second tuple element: triplets.shape[0]
  }
}

extern "C" void kernel_launch(void* const* d_in, const int* in_sizes, int n_in,
                              void* d_out, int out_size, void* d_ws, size_t ws_size,
                              hipStream_t stream) {
  const float* emb = (const float*)d_in[0];
  const long long* trip = (const long long*)d_in[1];
  float* out = (float*)d_out;
  float* losses = (float*)d_ws;

  const int T = in_sizes[1] / (3 * KPAIRS);   // 4096
  const int blocks = (T + 7) / 8;             // 8 waves (triplets) per 256-thread block

  lifted_loss_kernel<<<blocks, 256, 0, stream>>>(emb, trip, losses, T);
  lifted_loss_reduce<<<1, 256, 0, stream>>>(losses, out, T);
}